// residue_75926431859273
// MI455X (gfx1250) — compile-verified
//
#include <hip/hip_runtime.h>

typedef float v2f __attribute__((ext_vector_type(2)));
typedef float v8f __attribute__((ext_vector_type(8)));

#define LOG2E     1.4426950408889634f
#define NEG_2PI2  (-19.739208802178716f)   // -2*pi^2

// ---------------------------------------------------------------------------
// Setup: precompute per-(sym,atom) fractional positions xs[64][4], per-atom
// scaled ANISOU vectors uscl[16][8] (with -2*pi^2*log2(e) folded in, cross
// terms doubled), and occ / log2(occ).  Tiny; one block of 64 threads.
// ws layout (floats): [0..255] xs, [256..383] uscl, [384] occ, [385] log2occ
// ---------------------------------------------------------------------------
__global__ void residue_setup_kernel(const float* __restrict__ xyz,
                                     const float* __restrict__ u,
                                     const float* __restrict__ hidden_occ,
                                     const float* __restrict__ B_inv,
                                     const float* __restrict__ R,
                                     const float* __restrict__ T,
                                     float* __restrict__ ws) {
  const int t = threadIdx.x;
  if (t < 64) {
    const int s = t >> 4, a = t & 15;
    const float x = xyz[a * 3 + 0], y = xyz[a * 3 + 1], z = xyz[a * 3 + 2];
    const float f0 = B_inv[0] * x + B_inv[1] * y + B_inv[2] * z;
    const float f1 = B_inv[3] * x + B_inv[4] * y + B_inv[5] * z;
    const float f2 = B_inv[6] * x + B_inv[7] * y + B_inv[8] * z;
    const float* Rs = R + s * 9;
    const float* Ts = T + s * 3;
    ws[t * 4 + 0] = Rs[0] * f0 + Rs[1] * f1 + Rs[2] * f2 + Ts[0];
    ws[t * 4 + 1] = Rs[3] * f0 + Rs[4] * f1 + Rs[5] * f2 + Ts[1];
    ws[t * 4 + 2] = Rs[6] * f0 + Rs[7] * f1 + Rs[8] * f2 + Ts[2];
    ws[t * 4 + 3] = 0.0f;
  }
  if (t < 16) {
    const float* ua = u + t * 6;         // (u11,u22,u33,u12,u13,u23)
    const float c = NEG_2PI2 * LOG2E;    // so DW = exp2(qq . uscl)
    ws[256 + t * 8 + 0] = c * ua[0];
    ws[256 + t * 8 + 1] = c * ua[1];
    ws[256 + t * 8 + 2] = c * ua[2];
    ws[256 + t * 8 + 3] = 2.0f * c * ua[3];
    ws[256 + t * 8 + 4] = 2.0f * c * ua[4];
    ws[256 + t * 8 + 5] = 2.0f * c * ua[5];
    ws[256 + t * 8 + 6] = 0.0f;
    ws[256 + t * 8 + 7] = 0.0f;
  }
  if (t == 0) {
    const float xo = hidden_occ[0];
    const float occ = 1.0f / (1.0f + __expf(-xo));
    ws[384] = occ;
    ws[385] = __log2f(occ);
  }
}

// ---------------------------------------------------------------------------
// Per-wave tile body.  FULL=true: tile fully in range (no clamps, plain
// coalescible loads).  FULL=false: clamped tail (never taken for H%16==0).
// All ws reads are whole-float4 loads BEFORE any half-based select so the
// compiler cannot sink a component load under an exec-mask diamond.
// ---------------------------------------------------------------------------
template <bool FULL>
__device__ __forceinline__ void sf_tile(
    const int*   __restrict__ hkl,
    const float* __restrict__ sv,
    const float* __restrict__ sf,   // [A=16, H]
    const float* __restrict__ R,    // [4,3,3]
    const float* __restrict__ ws,
    float* __restrict__ out,        // [2, H]
    int H, int h0, int lane) {
  const int half = lane >> 4;       // 0: K0/K1 half, 1: K2/K3 half
  const int m    = lane & 15;       // reflection-in-tile (A) / atom (B,C)

  const float4* xs4 = (const float4*)ws;          // [64] xs entries
  const float4* us4 = (const float4*)(ws + 256);  // [16][2] uscl entries
  const float   l2occ = ws[385];

  int hm = h0 + m;
  if (!FULL) { if (hm > H - 1) hm = H - 1; }
  const float hx = (float)hkl[hm * 3 + 0];
  const float hy = (float)hkl[hm * 3 + 1];
  const float hz = (float)hkl[hm * 3 + 2];
  const float s0 = sv[hm * 3 + 0];
  const float s1 = sv[hm * 3 + 1];
  const float s2 = sv[hm * 3 + 2];

  // A (phase): 16x4, rows = (h,k,l,0)
  v2f pa;
  pa.x = half ? hz   : hx;   // K2 : K0
  pa.y = half ? 0.0f : hy;   // K3 : K1

  // B (DW exponent): scaled U per atom n=m -- two indivisible b128 loads.
  //   chain 1 K rows: (u11c, u22c, u33c, 2u12c)
  //   chain 2 K rows: (2u13c, 2u23c, log2occ, 0)  <- occ via spare K slot
  const float4 u0 = us4[m * 2 + 0];   // (u11c,u22c,u33c,2u12c)
  const float4 u1 = us4[m * 2 + 1];   // (2u13c,2u23c,0,0)
  v2f eb1, eb2;
  eb1.x = half ? u0.z  : u0.x;
  eb1.y = half ? u0.w  : u0.y;
  eb2.x = half ? l2occ : u1.x;
  eb2.y = half ? 0.0f  : u1.y;

  // xs entries for all 4 sym ops: indivisible b128 loads, hoisted.
  float4 xv[4];
  #pragma unroll
  for (int s = 0; s < 4; ++s) xv[s] = xs4[s * 16 + m];

  // scattering factors for C/D element (M = r + 8*half, N = m):
  // 8 consecutive floats from this atom's row.
  float sfv[8];
  {
    const int hb = h0 + 8 * half;
    const float* sfp = sf + (size_t)m * (size_t)H + hb;
    #pragma unroll
    for (int r = 0; r < 8; ++r) {
      if (FULL) {
        sfv[r] = sfp[r];
      } else {
        int hh = hb + r; if (hh > H - 1) hh = H - 1;
        sfv[r] = sf[(size_t)m * (size_t)H + hh];
      }
    }
  }

  float accR[8], accI[8];
  #pragma unroll
  for (int r = 0; r < 8; ++r) { accR[r] = 0.0f; accI[r] = 0.0f; }

  #pragma unroll
  for (int s = 0; s < 4; ++s) {
    // q = sv . R_s  (q_j = sum_i sv_i * R[s][i][j])
    const float* Rs = R + s * 9;
    const float q0 = s0 * Rs[0] + s1 * Rs[3] + s2 * Rs[6];
    const float q1 = s0 * Rs[1] + s1 * Rs[4] + s2 * Rs[7];
    const float q2 = s0 * Rs[2] + s1 * Rs[5] + s2 * Rs[8];

    // A (DW exponent): chain1 rows (q0^2,q1^2,q2^2,q0q1),
    //                  chain2 rows (q0q2,q1q2, 1.0, 0) -> 1.0*log2occ term
    v2f ea1, ea2;
    ea1.x = half ? q2 * q2 : q0 * q0;
    ea1.y = half ? q0 * q1 : q1 * q1;
    ea2.x = half ? 1.0f    : q0 * q2;
    ea2.y = half ? 0.0f    : q1 * q2;

    // B (phase): xs columns for atom n=m, sym s (already in registers)
    v2f pb;
    pb.x = half ? xv[s].z : xv[s].x;
    pb.y = half ? 0.0f    : xv[s].y;

    const v8f zc = {};
    v8f pT = __builtin_amdgcn_wmma_f32_16x16x4_f32(
        false, pa, false, pb, (short)0, zc, false, false);
    v8f e1 = __builtin_amdgcn_wmma_f32_16x16x4_f32(
        false, ea1, false, eb1, (short)0, zc, false, false);
    v8f e2 = __builtin_amdgcn_wmma_f32_16x16x4_f32(
        false, ea2, false, eb2, (short)0, e1, false, false);

    #pragma unroll
    for (int r = 0; r < 8; ++r) {
      const float amp = sfv[r] * __builtin_amdgcn_exp2f(e2[r]);
      // v_cos/v_sin compute cos/sin(2*pi * x): feed raw dot product
      accR[r] = fmaf(amp, __builtin_amdgcn_cosf(pT[r]), accR[r]);
      accI[r] = fmaf(amp, __builtin_amdgcn_sinf(pT[r]), accI[r]);
    }
  }

  // Reduce over atoms: butterfly within each 16-lane half-group.
  #pragma unroll
  for (int r = 0; r < 8; ++r) {
    #pragma unroll
    for (int st = 8; st >= 1; st >>= 1) {
      accR[r] += __shfl_xor(accR[r], st, 32);
      accI[r] += __shfl_xor(accI[r], st, 32);
    }
    const int hh = h0 + r + 8 * half;   // M = r + 8*half
    if (FULL) {
      if (m == 0) { out[hh] = accR[r]; out[H + hh] = accI[r]; }
    } else {
      if (m == 0 && hh < H) { out[hh] = accR[r]; out[H + hh] = accI[r]; }
    }
  }
}

// ---------------------------------------------------------------------------
__global__ __launch_bounds__(256) void residue_sf_kernel(
    const int*   __restrict__ hkl,
    const float* __restrict__ sv,
    const float* __restrict__ sf,
    const float* __restrict__ R,
    const float* __restrict__ ws,
    float* __restrict__ out,
    int H) {
  const int lane = threadIdx.x & 31;
  const int wave = threadIdx.x >> 5;
  const int h0   = (blockIdx.x * 8 + wave) * 16;
  if (h0 >= H) return;                       // wave-uniform
  if (h0 + 16 <= H) {                        // wave-uniform branch
    sf_tile<true >(hkl, sv, sf, R, ws, out, H, h0, lane);
  } else {
    sf_tile<false>(hkl, sv, sf, R, ws, out, H, h0, lane);
  }
}

// ---------------------------------------------------------------------------
extern "C" void kernel_launch(void* const* d_in, const int* in_sizes, int n_in,
                              void* d_out, int out_size, void* d_ws, size_t ws_size,
                              hipStream_t stream) {
  const int*   hkl   = (const int*)  d_in[0];
  const float* sv    = (const float*)d_in[1];
  const float* xyz   = (const float*)d_in[2];
  const float* u     = (const float*)d_in[3];
  const float* hocc  = (const float*)d_in[4];
  const float* sf    = (const float*)d_in[5];
  const float* B_inv = (const float*)d_in[6];
  const float* symR  = (const float*)d_in[7];
  const float* symT  = (const float*)d_in[8];
  float* out = (float*)d_out;
  float* ws  = (float*)d_ws;

  const int H = in_sizes[0] / 3;
  if (H <= 0) return;

  residue_setup_kernel<<<1, 64, 0, stream>>>(xyz, u, hocc, B_inv, symR, symT, ws);

  const int blocks = (H + 127) / 128;   // 8 waves/block * 16 reflections/wave
  residue_sf_kernel<<<blocks, 256, 0, stream>>>(hkl, sv, sf, symR, ws, out, H);
}